// RNN_9363028705535
// MI455X (gfx1250) — compile-verified
//
#include <hip/hip_runtime.h>
#include <math.h>

typedef __attribute__((ext_vector_type(2))) float v2f;
typedef __attribute__((ext_vector_type(8))) float v8f;

#define T_LEN 262144
#define ISZ   78
#define H     128
#define OUT   60

// ---------------------------------------------------------------------------
// Kernel 1: pre[T,128] = x[T,78] @ W_ih^T + (b_ih + b_hh)
// WMMA f32 16x16x4. Block = 256 threads = 8 waves; wave w owns column tile
// n0 = 16*w (covers all 128 hidden units); blockIdx.x owns 16 rows of time.
// ---------------------------------------------------------------------------
__global__ __launch_bounds__(256) void k_inproj(const float* __restrict__ x,
                                                const float* __restrict__ Wih,
                                                const float* __restrict__ bih,
                                                const float* __restrict__ bhh,
                                                float* __restrict__ pre) {
  const int lane = threadIdx.x & 31;
  const int wave = threadIdx.x >> 5;      // 0..7  -> column tile
  const int half = lane >> 4;             // 0/1   -> K sub-pair
  const int m    = lane & 15;
  const int r0   = blockIdx.x * 16;
  const int n    = wave * 16 + m;         // hidden unit (column)
  const int row  = r0 + m;                // time row for A fragment

  v8f acc = {};
  for (int k = 0; k < ISZ; k += 4) {      // K padded 78 -> 80, zero-fill tail
    const int ka = k + 2 * half;
    v2f a, b;
    a.x = (ka     < ISZ) ? x[row * ISZ + ka    ] : 0.f;
    a.y = (ka + 1 < ISZ) ? x[row * ISZ + ka + 1] : 0.f;
    b.x = (ka     < ISZ) ? Wih[n * ISZ + ka    ] : 0.f;   // B[k][n] = W_ih[n][k]
    b.y = (ka + 1 < ISZ) ? Wih[n * ISZ + ka + 1] : 0.f;
    acc = __builtin_amdgcn_wmma_f32_16x16x4_f32(false, a, false, b,
                                                (short)0, acc, false, false);
  }
  const float bias = bih[n] + bhh[n];
#pragma unroll
  for (int i = 0; i < 8; ++i) {
    const int rr = r0 + i + 8 * half;     // C layout: VGPR i -> row i + 8*half
    pre[rr * H + n] = acc[i] + bias;
  }
}

// ---------------------------------------------------------------------------
// Kernel 2: sequential scan  h_t = tanh(pre_t + W_hh h_{t-1});  P overwritten
// in place with h_t. One workgroup, 128 threads (4 waves). W_hh row j lives
// in 128 VGPRs of thread j; h double-buffered in LDS (broadcast reads);
// one barrier per step; next pre prefetched under the 128 FMAs.
// ---------------------------------------------------------------------------
__global__ __launch_bounds__(128) void k_recur(const float* __restrict__ Whh,
                                               float* __restrict__ P) {
  __shared__ float hbuf[2][H];
  const int j = threadIdx.x;

  float w[H];
#pragma unroll
  for (int k = 0; k < H; k += 4) {
    const float4 t4 = *(const float4*)&Whh[j * H + k];
    w[k] = t4.x; w[k + 1] = t4.y; w[k + 2] = t4.z; w[k + 3] = t4.w;
  }
  hbuf[0][j] = 0.f;
  __syncthreads();

  float p = P[j];                          // pre for t = 0
  for (int t = 0; t < T_LEN; ++t) {
    const float* hc = hbuf[t & 1];
    const float p_next = (t + 1 < T_LEN) ? P[(t + 1) * H + j] : 0.f;  // prefetch
    float a0 = 0.f, a1 = 0.f, a2 = 0.f, a3 = 0.f;
#pragma unroll
    for (int k = 0; k < H; k += 4) {       // 32 b128 LDS broadcasts, 128 FMAs
      const float4 hv = *(const float4*)&hc[k];
      a0 = fmaf(w[k    ], hv.x, a0);
      a1 = fmaf(w[k + 1], hv.y, a1);
      a2 = fmaf(w[k + 2], hv.z, a2);
      a3 = fmaf(w[k + 3], hv.w, a3);
    }
    const float hn = tanhf(p + ((a0 + a1) + (a2 + a3)));
    hbuf[(t + 1) & 1][j] = hn;             // write opposite buffer: no WAR race
    P[t * H + j] = hn;                     // stream h_t out (in place over pre)
    p = p_next;
    __syncthreads();                       // single barrier per step
  }
}

// ---------------------------------------------------------------------------
// Kernel 3: out = log_softmax(hs @ W_o^T + b_o). One wave per 16-row tile.
// WMMA f32 16x16x4, 4 column tiles (60 padded to 64), K = 128.
// Softmax via padded LDS staging; lanes 0..15 each own one row.
// ---------------------------------------------------------------------------
__global__ __launch_bounds__(32) void k_outproj(const float* __restrict__ P,
                                                const float* __restrict__ Wo,
                                                const float* __restrict__ bo,
                                                float* __restrict__ out) {
  __shared__ float tile[16][65];           // pad 64->65 to dodge bank conflicts
  const int lane = threadIdx.x & 31;
  const int half = lane >> 4;
  const int m    = lane & 15;
  const int r0   = blockIdx.x * 16;

  v8f acc[4] = {};
  for (int k = 0; k < H; k += 4) {
    const int ka = k + 2 * half;
    v2f a;
    a.x = P[(r0 + m) * H + ka];
    a.y = P[(r0 + m) * H + ka + 1];
#pragma unroll
    for (int c = 0; c < 4; ++c) {
      const int n = c * 16 + m;
      v2f b;
      b.x = (n < OUT) ? Wo[n * H + ka    ] : 0.f;
      b.y = (n < OUT) ? Wo[n * H + ka + 1] : 0.f;
      acc[c] = __builtin_amdgcn_wmma_f32_16x16x4_f32(false, a, false, b,
                                                     (short)0, acc[c], false, false);
    }
  }
#pragma unroll
  for (int c = 0; c < 4; ++c) {
    const int col  = c * 16 + m;
    const float bb = (col < OUT) ? bo[col] : 0.f;
#pragma unroll
    for (int i = 0; i < 8; ++i)
      tile[i + 8 * half][col] = acc[c][i] + bb;
  }
  __syncthreads();

  if (half == 0) {                          // lanes 0..15: one row each
    float mx = tile[m][0];
    for (int nn = 1; nn < OUT; ++nn) mx = fmaxf(mx, tile[m][nn]);
    float s = 0.f;
    for (int nn = 0; nn < OUT; ++nn) s += __expf(tile[m][nn] - mx);
    const float lse = mx + __logf(s);
    for (int nn = 0; nn < OUT; ++nn)
      out[(r0 + m) * OUT + nn] = tile[m][nn] - lse;
  }
}

// ---------------------------------------------------------------------------
extern "C" void kernel_launch(void* const* d_in, const int* in_sizes, int n_in,
                              void* d_out, int out_size, void* d_ws, size_t ws_size,
                              hipStream_t stream) {
  (void)in_sizes; (void)n_in; (void)out_size; (void)ws_size;
  const float* x   = (const float*)d_in[0];
  const float* Wih = (const float*)d_in[1];
  const float* Whh = (const float*)d_in[2];
  const float* bih = (const float*)d_in[3];
  const float* bhh = (const float*)d_in[4];
  const float* Wo  = (const float*)d_in[5];
  const float* bo  = (const float*)d_in[6];
  float* out = (float*)d_out;
  float* P   = (float*)d_ws;               // T*H f32 = 128 MB scratch

  k_inproj <<<T_LEN / 16, 256, 0, stream>>>(x, Wih, bih, bhh, P);
  k_recur  <<<1,          128, 0, stream>>>(Whh, P);
  k_outproj<<<T_LEN / 16,  32, 0, stream>>>(P, Wo, bo, out);
}